// MambaAdapter_37503654429369
// MI455X (gfx1250) — compile-verified
//
#include <hip/hip_runtime.h>
#include <hip/hip_bf16.h>

typedef __attribute__((ext_vector_type(2))) float v2f;
typedef __attribute__((ext_vector_type(4))) float v4f;
typedef __attribute__((ext_vector_type(8))) float v8f;

#define L_SEQ   8192
#define DM      256
#define DI      512
#define DS      16
#define DTR     16
#define NXZ     1024   // 2*DI
#define NDBC    48     // DTR + 2*DS
#define NCHUNK  32
#define CHUNK   256    // L_SEQ / NCHUNK

// ---------------------------------------------------------------------------
// K1: xz[l, n] = sum_c x[c*L + l] * W_in[c*NXZ + n]      (A is K-major in mem)
// grid (128, 8), block 128 (4 waves). Wave computes a 64x32 tile (4x2 WMMA
// sub-tiles, 8 v8f accumulators = 64 VGPRs -> no spill). 12 loads / 8 WMMAs.
// ---------------------------------------------------------------------------
__global__ void __launch_bounds__(128, 1)
k_gemm_in(const float* __restrict__ x,
          const float* __restrict__ W_in,
          float* __restrict__ xz) {
  const int wave = threadIdx.x >> 5;
  const int lane = threadIdx.x & 31;
  const int lm = lane & 15, lh = lane >> 4;
  const int m0 = blockIdx.x * 64;
  const int n0 = blockIdx.y * 128 + wave * 32;
  v8f acc[4][2] = {};
  for (int k0 = 0; k0 < DM; k0 += 4) {
    const int ka = k0 + 2 * lh;
    v2f a[4], b[2];
    const float* xa0 = x + (size_t)ka * L_SEQ + m0 + lm;          // k = ka
    const float* xa1 = xa0 + L_SEQ;                               // k = ka+1
#pragma unroll
    for (int mi = 0; mi < 4; ++mi) { a[mi].x = xa0[16 * mi]; a[mi].y = xa1[16 * mi]; }
    const float* wb0 = W_in + (size_t)ka * NXZ + n0 + lm;
    const float* wb1 = wb0 + NXZ;
#pragma unroll
    for (int ni = 0; ni < 2; ++ni) { b[ni].x = wb0[16 * ni]; b[ni].y = wb1[16 * ni]; }
#pragma unroll
    for (int mi = 0; mi < 4; ++mi)
#pragma unroll
      for (int ni = 0; ni < 2; ++ni)
        acc[mi][ni] = __builtin_amdgcn_wmma_f32_16x16x4_f32(
            false, a[mi], false, b[ni], (short)0, acc[mi][ni], false, false);
  }
#pragma unroll
  for (int mi = 0; mi < 4; ++mi)
#pragma unroll
    for (int r = 0; r < 8; ++r) {
      const size_t row = (size_t)(m0 + 16 * mi + r + 8 * lh) * NXZ + n0 + lm;
#pragma unroll
      for (int ni = 0; ni < 2; ++ni)
        xz[row + 16 * ni] = acc[mi][ni][r];
    }
}

// ---------------------------------------------------------------------------
// K2: depthwise causal conv(4) + bias + SiLU over the first DI columns of xz.
// ---------------------------------------------------------------------------
__global__ void k_conv_silu(const float* __restrict__ xz,
                            const float* __restrict__ conv_w,
                            const float* __restrict__ conv_b,
                            float* __restrict__ xi) {
  const int idx = blockIdx.x * 256 + threadIdx.x;   // over L*DI
  if (idx >= L_SEQ * DI) return;
  const int d = idx & (DI - 1);
  const int l = idx >> 9;
  float acc = conv_b[d];
#pragma unroll
  for (int j = 0; j < 4; ++j) {
    const int ls = l - 3 + j;
    const float v = (ls >= 0) ? xz[(size_t)ls * NXZ + d] : 0.f;
    acc = fmaf(conv_w[d * 4 + j], v, acc);
  }
  xi[idx] = acc / (1.f + __expf(-acc));
}

// ---------------------------------------------------------------------------
// K3: dbc[l, n] = sum_d xi[l, d] * W_xproj[d*48 + n], n < 48.
// grid 256, block 32. Wave covers 32 rows x all 3 N-tiles (6 accumulators);
// A fragments are contiguous pairs -> b64 loads, reused across 3 N-tiles.
// ---------------------------------------------------------------------------
__global__ void __launch_bounds__(32, 1)
k_gemm_xproj(const float* __restrict__ xi,
             const float* __restrict__ W_xproj,
             float* __restrict__ dbc) {
  const int lane = threadIdx.x & 31;
  const int lm = lane & 15, lh = lane >> 4;
  const int m0 = blockIdx.x * 32;
  v8f acc[2][3] = {};
  for (int k0 = 0; k0 < DI; k0 += 4) {
    const int ka = k0 + 2 * lh;
    v2f a[2], b[3];
#pragma unroll
    for (int mi = 0; mi < 2; ++mi)
      a[mi] = *(const v2f*)(xi + (size_t)(m0 + 16 * mi + lm) * DI + ka);
    const float* wb0 = W_xproj + (size_t)ka * NDBC + lm;
    const float* wb1 = wb0 + NDBC;
#pragma unroll
    for (int ni = 0; ni < 3; ++ni) { b[ni].x = wb0[16 * ni]; b[ni].y = wb1[16 * ni]; }
#pragma unroll
    for (int mi = 0; mi < 2; ++mi)
#pragma unroll
      for (int ni = 0; ni < 3; ++ni)
        acc[mi][ni] = __builtin_amdgcn_wmma_f32_16x16x4_f32(
            false, a[mi], false, b[ni], (short)0, acc[mi][ni], false, false);
  }
#pragma unroll
  for (int mi = 0; mi < 2; ++mi)
#pragma unroll
    for (int r = 0; r < 8; ++r) {
      const size_t row = (size_t)(m0 + 16 * mi + r + 8 * lh) * NDBC + lm;
#pragma unroll
      for (int ni = 0; ni < 3; ++ni)
        dbc[row + 16 * ni] = acc[mi][ni][r];
    }
}

// ---------------------------------------------------------------------------
// K4: dt[l, d] = softplus( sum_r dbc[l*48 + r] * W_dt[r*DI + d] + b_dt[d] )
// grid (128, 8), block 128. Wave: 64 rows x one 16-wide N tile, K=16.
// ---------------------------------------------------------------------------
__global__ void __launch_bounds__(128, 1)
k_gemm_dt(const float* __restrict__ dbc,
          const float* __restrict__ W_dt,
          const float* __restrict__ b_dt,
          float* __restrict__ dt) {
  const int wave = threadIdx.x >> 5;
  const int lane = threadIdx.x & 31;
  const int lm = lane & 15, lh = lane >> 4;
  const int m0 = blockIdx.x * 64;
  const int n0 = blockIdx.y * 64 + wave * 16;
  v8f acc[4] = {};
#pragma unroll
  for (int k0 = 0; k0 < DTR; k0 += 4) {
    const int ka = k0 + 2 * lh;
    v2f a[4], b;
#pragma unroll
    for (int mi = 0; mi < 4; ++mi)
      a[mi] = *(const v2f*)(dbc + (size_t)(m0 + 16 * mi + lm) * NDBC + ka);
    b.x = W_dt[(ka + 0) * DI + n0 + lm];
    b.y = W_dt[(ka + 1) * DI + n0 + lm];
#pragma unroll
    for (int mi = 0; mi < 4; ++mi)
      acc[mi] = __builtin_amdgcn_wmma_f32_16x16x4_f32(
          false, a[mi], false, b, (short)0, acc[mi], false, false);
  }
  const float bias = b_dt[n0 + lm];
#pragma unroll
  for (int mi = 0; mi < 4; ++mi)
#pragma unroll
    for (int r = 0; r < 8; ++r) {
      float v = acc[mi][r] + bias;
      v = (v > 20.f) ? v : log1pf(__expf(v));   // softplus
      dt[(size_t)(m0 + 16 * mi + r + 8 * lh) * DI + n0 + lm] = v;
    }
}

// ---------------------------------------------------------------------------
// K5a: chunked scan pass 1 — per (chunk, d, s): prod(dA) and local h (h0 = 0).
// grid (NCHUNK, DI/16), block 256 = 16 d x 16 s.
// ---------------------------------------------------------------------------
__global__ void k_scan_pass1(const float* __restrict__ dt,
                             const float* __restrict__ dbc,
                             const float* __restrict__ xi,
                             const float* __restrict__ A_log,
                             float* __restrict__ chunk_prodA,
                             float* __restrict__ chunk_hend) {
  const int s = threadIdx.x & 15;
  const int dl = threadIdx.x >> 4;
  const int d = blockIdx.y * 16 + dl;
  const int c = blockIdx.x;
  const float Aval = -__expf(A_log[d * DS + s]);
  float h = 0.f, prodA = 1.f;
  const int l0 = c * CHUNK;
  for (int t = 0; t < CHUNK; ++t) {
    const int l = l0 + t;
    const float dtv = dt[(size_t)l * DI + d];
    const float Bv  = dbc[(size_t)l * NDBC + DTR + s];
    const float xv  = xi[(size_t)l * DI + d];
    const float dA  = __expf(dtv * Aval);
    h = fmaf(dA, h, dtv * Bv * xv);
    prodA *= dA;
  }
  const size_t o = (size_t)c * (DI * DS) + d * DS + s;
  chunk_prodA[o] = prodA;
  chunk_hend[o]  = h;
}

// ---------------------------------------------------------------------------
// K5b: combine chunk summaries sequentially (32 steps per (d, s) lane).
// ---------------------------------------------------------------------------
__global__ void k_scan_combine(const float* __restrict__ chunk_prodA,
                               const float* __restrict__ chunk_hend,
                               float* __restrict__ chunk_hin) {
  const int i = blockIdx.x * 256 + threadIdx.x;   // (d, s) flat, < 8192
  float h = 0.f;
#pragma unroll 4
  for (int c = 0; c < NCHUNK; ++c) {
    const size_t o = (size_t)c * (DI * DS) + i;
    chunk_hin[o] = h;
    h = fmaf(chunk_prodA[o], h, chunk_hend[o]);
  }
}

// ---------------------------------------------------------------------------
// K5c: scan pass 2 — re-derive dA/dBx with correct incoming h, reduce over s,
// apply D_skip and SiLU(z) gating. y[l, d] output.
// ---------------------------------------------------------------------------
__global__ void k_scan_pass2(const float* __restrict__ dt,
                             const float* __restrict__ dbc,
                             const float* __restrict__ xi,
                             const float* __restrict__ xz,
                             const float* __restrict__ A_log,
                             const float* __restrict__ D_skip,
                             const float* __restrict__ chunk_hin,
                             float* __restrict__ y) {
  const int s = threadIdx.x & 15;
  const int dl = threadIdx.x >> 4;
  const int d = blockIdx.y * 16 + dl;
  const int c = blockIdx.x;
  const float Aval = -__expf(A_log[d * DS + s]);
  const float dsk = D_skip[d];
  float h = chunk_hin[(size_t)c * (DI * DS) + d * DS + s];
  const int l0 = c * CHUNK;
  for (int t = 0; t < CHUNK; ++t) {
    const int l = l0 + t;
    const float dtv = dt[(size_t)l * DI + d];
    const float Bv  = dbc[(size_t)l * NDBC + DTR + s];
    const float Cv  = dbc[(size_t)l * NDBC + DTR + DS + s];
    const float xv  = xi[(size_t)l * DI + d];
    const float dA  = __expf(dtv * Aval);
    h = fmaf(dA, h, dtv * Bv * xv);
    float contrib = h * Cv;
    contrib += __shfl_xor(contrib, 1);
    contrib += __shfl_xor(contrib, 2);
    contrib += __shfl_xor(contrib, 4);
    contrib += __shfl_xor(contrib, 8);
    if (s == 0) {
      const float zv = xz[(size_t)l * NXZ + DI + d];
      const float gate = zv / (1.f + __expf(-zv));
      y[(size_t)l * DI + d] = (contrib + xv * dsk) * gate;
    }
  }
}

// ---------------------------------------------------------------------------
// K6: out[c*L + l] = sum_d y[l, d] * W_out[d*DM + c]   (transposed store)
// grid 128 (1-D), block 128. Wave tile 64x32 (4x2 sub-tiles).
// A fragments are contiguous b64 loads; output written as b128 pairs.
// ---------------------------------------------------------------------------
__global__ void __launch_bounds__(128, 1)
k_gemm_out(const float* __restrict__ y,
           const float* __restrict__ W_out,
           float* __restrict__ out) {
  const int wave = threadIdx.x >> 5;
  const int lane = threadIdx.x & 31;
  const int lm = lane & 15, lh = lane >> 4;
  const int m0 = blockIdx.x * 64;                 // l tile
  const int n0 = blockIdx.y * 128 + wave * 32;    // c tile
  v8f acc[4][2] = {};
  for (int k0 = 0; k0 < DI; k0 += 4) {
    const int ka = k0 + 2 * lh;
    v2f a[4], b[2];
#pragma unroll
    for (int mi = 0; mi < 4; ++mi)
      a[mi] = *(const v2f*)(y + (size_t)(m0 + 16 * mi + lm) * DI + ka);
    const float* wb0 = W_out + (size_t)ka * DM + n0 + lm;
    const float* wb1 = wb0 + DM;
#pragma unroll
    for (int ni = 0; ni < 2; ++ni) { b[ni].x = wb0[16 * ni]; b[ni].y = wb1[16 * ni]; }
#pragma unroll
    for (int mi = 0; mi < 4; ++mi)
#pragma unroll
      for (int ni = 0; ni < 2; ++ni)
        acc[mi][ni] = __builtin_amdgcn_wmma_f32_16x16x4_f32(
            false, a[mi], false, b[ni], (short)0, acc[mi][ni], false, false);
  }
#pragma unroll
  for (int ni = 0; ni < 2; ++ni) {
    const size_t cbase = (size_t)(n0 + 16 * ni + lm) * L_SEQ;
#pragma unroll
    for (int mi = 0; mi < 4; ++mi) {
      const int l0 = m0 + 16 * mi + 8 * lh;       // 8 consecutive l per lane
      v4f lo = { acc[mi][ni][0], acc[mi][ni][1], acc[mi][ni][2], acc[mi][ni][3] };
      v4f hi = { acc[mi][ni][4], acc[mi][ni][5], acc[mi][ni][6], acc[mi][ni][7] };
      *(v4f*)(out + cbase + l0 + 0) = lo;
      *(v4f*)(out + cbase + l0 + 4) = hi;
    }
  }
}

// ---------------------------------------------------------------------------
extern "C" void kernel_launch(void* const* d_in, const int* in_sizes, int n_in,
                              void* d_out, int out_size, void* d_ws, size_t ws_size,
                              hipStream_t stream) {
  (void)in_sizes; (void)n_in; (void)out_size; (void)ws_size;
  const float* x      = (const float*)d_in[0];
  const float* W_in   = (const float*)d_in[1];
  const float* conv_w = (const float*)d_in[2];
  const float* conv_b = (const float*)d_in[3];
  const float* W_xprj = (const float*)d_in[4];
  const float* W_dt   = (const float*)d_in[5];
  const float* b_dt   = (const float*)d_in[6];
  const float* A_log  = (const float*)d_in[7];
  const float* D_skip = (const float*)d_in[8];
  const float* W_out  = (const float*)d_in[9];
  float* out = (float*)d_out;

  float* ws  = (float*)d_ws;
  float* xz  = ws;                          // L * 1024
  float* xi  = xz  + (size_t)L_SEQ * NXZ;   // L * 512
  float* dbc = xi  + (size_t)L_SEQ * DI;    // L * 48
  float* dt  = dbc + (size_t)L_SEQ * NDBC;  // L * 512
  float* y   = dt  + (size_t)L_SEQ * DI;    // L * 512
  float* pA  = y   + (size_t)L_SEQ * DI;    // NCHUNK * 8192
  float* hE  = pA  + (size_t)NCHUNK * DI * DS;
  float* hI  = hE  + (size_t)NCHUNK * DI * DS;

  k_gemm_in   <<<dim3(L_SEQ / 64, NXZ / 128), 128, 0, stream>>>(x, W_in, xz);
  k_conv_silu <<<(L_SEQ * DI) / 256, 256, 0, stream>>>(xz, conv_w, conv_b, xi);
  k_gemm_xproj<<<L_SEQ / 32, 32, 0, stream>>>(xi, W_xprj, dbc);
  k_gemm_dt   <<<dim3(L_SEQ / 64, DI / 64), 128, 0, stream>>>(dbc, W_dt, b_dt, dt);
  k_scan_pass1<<<dim3(NCHUNK, DI / 16), 256, 0, stream>>>(dt, dbc, xi, A_log, pA, hE);
  k_scan_combine<<<32, 256, 0, stream>>>(pA, hE, hI);
  k_scan_pass2<<<dim3(NCHUNK, DI / 16), 256, 0, stream>>>(dt, dbc, xi, xz, A_log,
                                                          D_skip, hI, y);
  k_gemm_out  <<<L_SEQ / 64, 128, 0, stream>>>(y, W_out, out);
}